// AgentNetworkSeparate_67233418051917
// MI455X (gfx1250) — compile-verified
//
#include <hip/hip_runtime.h>
#include <hip/hip_bf16.h>

typedef __attribute__((ext_vector_type(2))) float v2f;
typedef __attribute__((ext_vector_type(4))) float v4f;
typedef __attribute__((ext_vector_type(8))) float v8f;

#define T_TOK   32768
#define FEAT_N  136
#define NEXP    8
#define ROWS    32                            // M-rows per block tile
#define TPAD    (T_TOK + ROWS * NEXP)         // padded row capacity (33024)
#define MAXTILE (T_TOK / ROWS + NEXP)         // 1032 blocks upper bound

// meta[] int layout
#define M_CNT    0   // counts[8]
#define M_CUR    8   // cursors[8]
#define M_SEG    16  // padded segment starts[8]
#define M_TSTART 24  // tile prefix[9]
#define M_TOTAL  33  // total tiles

__global__ __launch_bounds__(256) void moe_init(int* __restrict__ perm, int* __restrict__ meta) {
    int i = blockIdx.x * blockDim.x + threadIdx.x;
    if (i < TPAD) perm[i] = -1;
    if (i < 16)   meta[i] = 0;     // counts + cursors
}

__global__ __launch_bounds__(256) void moe_ids(const float* __restrict__ input,
                                               int* __restrict__ ids, int* __restrict__ meta) {
    int t = blockIdx.x * blockDim.x + threadIdx.x;
    if (t >= T_TOK) return;
    const float* p = input + (size_t)t * FEAT_N + 128;
    float best = p[0];
    int bi = 0;
#pragma unroll
    for (int i = 1; i < NEXP; ++i) {
        float v = p[i];
        if (v > best) { best = v; bi = i; }   // strict > keeps first max (jnp.argmax)
    }
    ids[t] = bi;
    atomicAdd(&meta[M_CNT + bi], 1);
}

__global__ void moe_setup(int* __restrict__ meta) {
    if (threadIdx.x == 0 && blockIdx.x == 0) {
        int off = 0, toff = 0;
        for (int e = 0; e < NEXP; ++e) {
            meta[M_SEG + e]    = off;
            meta[M_TSTART + e] = toff;
            int tiles = (meta[M_CNT + e] + ROWS - 1) / ROWS;
            off  += tiles * ROWS;
            toff += tiles;
        }
        meta[M_TSTART + NEXP] = toff;
        meta[M_TOTAL] = toff;
    }
}

__global__ __launch_bounds__(256) void moe_scatter(const int* __restrict__ ids,
                                                   int* __restrict__ meta, int* __restrict__ perm) {
    int t = blockIdx.x * blockDim.x + threadIdx.x;
    if (t >= T_TOK) return;
    int e = ids[t];
    int slot = meta[M_SEG + e] + atomicAdd(&meta[M_CUR + e], 1);
    perm[slot] = t;
}

// MODE: 0 = read sorted buffer / write sorted buffer
//       1 = gather rows from raw input via perm (layer 1)
//       2 = scatter rows to d_out via perm (layer 7)
template <int O, int D, bool RELU, int MODE>
__global__ __launch_bounds__(256) void expert_gemm(const float* __restrict__ in,
                                                   const float* __restrict__ W,
                                                   const float* __restrict__ B,
                                                   float* __restrict__ out,
                                                   const int* __restrict__ meta,
                                                   const int* __restrict__ perm,
                                                   const float* __restrict__ raw_input) {
    constexpr int LDSS = D + 4;                  // +4 floats: conflict-free row stride
    __shared__ float lds[ROWS * LDSS];

    const int tile = blockIdx.x;
    if (tile >= meta[M_TOTAL]) return;

    const int* tstart = meta + M_TSTART;
    int e = 0;
    while (e < NEXP - 1 && tile >= tstart[e + 1]) ++e;
    const int rowBase = meta[M_SEG + e] + (tile - tstart[e]) * ROWS;

    // Stage ROWS x D activation tile into LDS as float4 (b128 in / b128 out)
    constexpr int DV = D / 4;
    for (int idx = threadIdx.x; idx < ROWS * DV; idx += 256) {
        int r  = idx / DV;                        // powers of two -> shifts
        int c4 = (idx & (DV - 1)) * 4;
        v4f v;
        if (MODE == 1) {
            int tok = perm[rowBase + r];
            if (tok >= 0) v = *(const v4f*)(raw_input + (size_t)tok * FEAT_N + c4);
            else          { v[0] = 0.f; v[1] = 0.f; v[2] = 0.f; v[3] = 0.f; }
        } else {
            v = *(const v4f*)(in + (size_t)(rowBase + r) * 512 + c4);
        }
        *(v4f*)(lds + r * LDSS + c4) = v;
    }
    __syncthreads();

    const int lane = threadIdx.x & 31;
    const int wave = threadIdx.x >> 5;
    const int hi   = lane >> 4;          // 0|1 -> K pair select
    const int ln   = lane & 15;          // M for A, N for B/C/D
    const int koff = hi * 2;

    constexpr int NCT = O / 16;          // column tiles total
    constexpr int NT  = (NCT + 7) / 8;   // column tiles per wave
    v8f acc[NT][2];                      // [col tile][row half]
#pragma unroll
    for (int t = 0; t < NT; ++t)
#pragma unroll
        for (int h = 0; h < 2; ++h)
#pragma unroll
            for (int j = 0; j < 8; ++j) acc[t][h][j] = 0.0f;

    const float* ldsA0 = lds + ln * LDSS + koff;          // rows 0..15
    const float* ldsA1 = lds + (16 + ln) * LDSS + koff;   // rows 16..31

#pragma unroll 2
    for (int k = 0; k < D; k += 4) {
        v2f a0 = *(const v2f*)(ldsA0 + k);   // A 16x4 frag: lane<16 K0/K1, lane>=16 K2/K3
        v2f a1 = *(const v2f*)(ldsA1 + k);
#pragma unroll
        for (int t = 0; t < NT; ++t) {
            int ct = wave + t * 8;           // wave-uniform -> EXEC stays all-ones
            if (NCT < 8 && ct >= NCT) continue;
            const float* wp = W + ((size_t)e * O + (size_t)(ct * 16 + ln)) * D + k + koff;
            v2f b = *(const v2f*)wp;         // B 4x16 frag, reused by both row halves
            acc[t][0] = __builtin_amdgcn_wmma_f32_16x16x4_f32(
                            false, a0, false, b, (short)0, acc[t][0], false, false);
            acc[t][1] = __builtin_amdgcn_wmma_f32_16x16x4_f32(
                            false, a1, false, b, (short)0, acc[t][1], false, false);
        }
    }

    // Epilogue: bias (+ReLU), store. C/D layout: VGPR j -> M = j + 8*hi (+16 for half 1), N = ln.
#pragma unroll
    for (int t = 0; t < NT; ++t) {
        int ct = wave + t * 8;
        if (NCT < 8 && ct >= NCT) continue;
        int col = ct * 16 + ln;
        float bias = B[e * O + col];
        if (MODE == 2) {
#pragma unroll
            for (int h = 0; h < 2; ++h)
#pragma unroll
                for (int j = 0; j < 8; ++j) {
                    int m = h * 16 + hi * 8 + j;
                    float v = acc[t][h][j] + bias;
                    if (RELU) v = fmaxf(v, 0.0f);
                    int tok = perm[rowBase + m];
                    if (tok >= 0) out[(size_t)tok * O + col] = v;
                }
        } else {
            // one base pointer; row steps become immediate store offsets (j*512 floats)
            float* op = out + (size_t)(rowBase + hi * 8) * 512 + col;
#pragma unroll
            for (int h = 0; h < 2; ++h) {
                float* oph = op + (size_t)h * 16 * 512;
#pragma unroll
                for (int j = 0; j < 8; ++j) {
                    float v = acc[t][h][j] + bias;
                    if (RELU) v = fmaxf(v, 0.0f);
                    oph[j * 512] = v;
                }
            }
        }
    }
}

extern "C" void kernel_launch(void* const* d_in, const int* in_sizes, int n_in,
                              void* d_out, int out_size, void* d_ws, size_t ws_size,
                              hipStream_t stream) {
    (void)in_sizes; (void)n_in; (void)out_size; (void)ws_size;
    const float* input = (const float*)d_in[0];
    const float* W[7];
    const float* b[7];
    for (int i = 0; i < 7; ++i) {
        W[i] = (const float*)d_in[1 + 2 * i];
        b[i] = (const float*)d_in[2 + 2 * i];
    }
    float* outp = (float*)d_out;

    char* ws = (char*)d_ws;
    int*   ids  = (int*)ws;                              // 32768 ints
    int*   perm = (int*)(ws + 131072);                   // TPAD ints
    int*   meta = (int*)(ws + 266240);                   // 34 ints
    float* bufA = (float*)(ws + (1u << 20));             // TPAD x 512 f32 (~67.6 MB)
    float* bufB = (float*)(ws + 73400320u);              // TPAD x 512 f32

    moe_init<<<(TPAD + 255) / 256, 256, 0, stream>>>(perm, meta);
    moe_ids<<<T_TOK / 256, 256, 0, stream>>>(input, ids, meta);
    moe_setup<<<1, 32, 0, stream>>>(meta);
    moe_scatter<<<T_TOK / 256, 256, 0, stream>>>(ids, meta, perm);

    dim3 g(MAXTILE), blk(256);
    // L1: 128 -> 512, relu, gather from raw input
    expert_gemm<512, 128, true, 1><<<g, blk, 0, stream>>>(nullptr, W[0], b[0], bufA, meta, perm, input);
    // L2: 512 -> 512, relu
    expert_gemm<512, 512, true, 0><<<g, blk, 0, stream>>>(bufA, W[1], b[1], bufB, meta, perm, nullptr);
    // L3: 512 -> 512, no relu
    expert_gemm<512, 512, false, 0><<<g, blk, 0, stream>>>(bufB, W[2], b[2], bufA, meta, perm, nullptr);
    // L4: 512 -> 512, relu
    expert_gemm<512, 512, true, 0><<<g, blk, 0, stream>>>(bufA, W[3], b[3], bufB, meta, perm, nullptr);
    // L5: 512 -> 512, relu
    expert_gemm<512, 512, true, 0><<<g, blk, 0, stream>>>(bufB, W[4], b[4], bufA, meta, perm, nullptr);
    // L6: 512 -> 256, no relu
    expert_gemm<256, 512, false, 0><<<g, blk, 0, stream>>>(bufA, W[5], b[5], bufB, meta, perm, nullptr);
    // L7: 256 -> 32, no relu, scatter to d_out by token
    expert_gemm<32, 256, false, 2><<<g, blk, 0, stream>>>(bufB, W[6], b[6], outp, meta, perm, nullptr);
}